// EdgeConv_76158360092793
// MI455X (gfx1250) — compile-verified
//
#include <hip/hip_runtime.h>

typedef float v2f __attribute__((ext_vector_type(2)));
typedef float v8f __attribute__((ext_vector_type(8)));

#define B_ 8
#define N_ 2048
#define C_ 64
#define K_ 32
#define O1_ 64
#define O2_ 128
#define NEG_SLOPE 0.01f
#define RSQ_BN 0.99999500003749968f /* 1/sqrt(1+1e-5) */
#define NEG_INF (-3.402823466e38f)
#define XSTR 68 /* padded row stride (floats) for 32x64 LDS tiles */

// ---------------------------------------------------------------------------
// Kernel 1: exact kNN (top-32 by neg_dist = 2*inner - |xi|^2 - |xj|^2).
// One wave per point. Candidates parked in LDS (stride 32 -> conflict free),
// 32 rounds of shuffle-xor argmax with tie-break on lower index (top_k order).
// ---------------------------------------------------------------------------
__global__ __launch_bounds__(128) void knn_topk_kernel(const float* __restrict__ x,
                                                       int* __restrict__ idx) {
  __shared__ float cand[4][N_];
  const int lane = threadIdx.x & 31;
  const int wave = threadIdx.x >> 5;
  const int p = blockIdx.x * 4 + wave;
  const int b = p >> 11;
  const int i = p & (N_ - 1);
  const float* xb = x + (size_t)b * N_ * C_;

  float4 xi[16];
  const float4* xi4 = (const float4*)(xb + i * C_);
#pragma unroll
  for (int c = 0; c < 16; ++c) xi[c] = xi4[c];
  float sqi = 0.f;
#pragma unroll
  for (int c = 0; c < 16; ++c)
    sqi += xi[c].x * xi[c].x + xi[c].y * xi[c].y + xi[c].z * xi[c].z + xi[c].w * xi[c].w;

  float* my = cand[wave];
  for (int chunk = 0; chunk < N_ / 32; ++chunk) {
    const int j = chunk * 32 + lane;
    const float4* xj4 = (const float4*)(xb + j * C_);
    float inner = 0.f, sqj = 0.f;
#pragma unroll
    for (int c = 0; c < 16; ++c) {
      float4 q = xj4[c];
      inner += xi[c].x * q.x + xi[c].y * q.y + xi[c].z * q.z + xi[c].w * q.w;
      sqj   += q.x * q.x + q.y * q.y + q.z * q.z + q.w * q.w;
    }
    my[chunk * 32 + lane] = 2.f * inner - sqi - sqj;
  }

  // per-lane local max over its 64 candidates (entries lane + 32*c)
  float m = NEG_INF; int mc = 0;
  for (int c = 0; c < 64; ++c) {
    float v = my[c * 32 + lane];
    if (v > m) { m = v; mc = c; }
  }

  int keepj = 0;
  for (int t = 0; t < K_; ++t) {
    float v = m;
    int j = mc * 32 + lane;
#pragma unroll
    for (int off = 16; off >= 1; off >>= 1) {
      float vo = __shfl_xor(v, off, 32);
      int   jo = __shfl_xor(j, off, 32);
      if (vo > v || (vo == v && jo < j)) { v = vo; j = jo; }
    }
    if (lane == t) keepj = j;
    if ((j & 31) == lane) {  // winning lane: invalidate + rescan
      my[(j >> 5) * 32 + lane] = NEG_INF;
      m = NEG_INF; mc = 0;
      for (int c = 0; c < 64; ++c) {
        float vv = my[c * 32 + lane];
        if (vv > m) { m = vv; mc = c; }
      }
    }
  }
  idx[(size_t)p * K_ + lane] = keepj;
}

// ---------------------------------------------------------------------------
// Kernel 2: fused gather -> MLP(128->64) -> MLP(64->128) -> max over k,
// all matmuls on V_WMMA_F32_16X16X4_F32 (fp32-exact). One wave per point.
// ---------------------------------------------------------------------------
__global__ __launch_bounds__(128) void edgeconv_wmma_kernel(
    const float* __restrict__ x, const int* __restrict__ idx,
    const float* __restrict__ W1, const float* __restrict__ g1, const float* __restrict__ bb1,
    const float* __restrict__ W2, const float* __restrict__ g2, const float* __restrict__ bb2,
    float* __restrict__ out) {
  __shared__ v2f W1P[64 * 64];    // [c2][o]: (W1[o][2c2], W1[o][2c2+1])  32KB
  __shared__ v2f W2P[32 * 128];   // [c2][o]: (W2[o][2c2], W2[o][2c2+1])  32KB
  __shared__ float s1s[O1_], b1s[O1_], s2s[O2_], b2s[O2_];
  __shared__ __attribute__((aligned(16))) float XN[4][32 * XSTR];  // x_j - x_i
  __shared__ __attribute__((aligned(16))) float H1[4][32 * XSTR];  // layer-1 out

  const int tid = threadIdx.x;
  const int lane = tid & 31;
  const int wave = tid >> 5;
  const int lo = lane & 15;
  const int hi = lane >> 4;

  // Stage pair-packed transposed weights + BN params into LDS once per block.
  for (int e = tid; e < 64 * 64; e += 128) {
    int c2 = e >> 6, o = e & 63;
    v2f w; w.x = W1[o * (2 * C_) + 2 * c2]; w.y = W1[o * (2 * C_) + 2 * c2 + 1];
    W1P[e] = w;
  }
  for (int e = tid; e < 32 * 128; e += 128) {
    int c2 = e >> 7, o = e & 127;
    v2f w; w.x = W2[o * O1_ + 2 * c2]; w.y = W2[o * O1_ + 2 * c2 + 1];
    W2P[e] = w;
  }
  if (tid < O1_) { s1s[tid] = g1[tid] * RSQ_BN; b1s[tid] = bb1[tid]; }
  s2s[tid] = g2[tid] * RSQ_BN;
  b2s[tid] = bb2[tid];
  __syncthreads();

  const int gw = blockIdx.x * 4 + wave;
  const int nwaves = gridDim.x * 4;
  for (int p = gw; p < B_ * N_; p += nwaves) {
    const int b = p >> 11;
    const int i = p & (N_ - 1);
    const float* xb = x + (size_t)b * N_ * C_;

    float4 xi[16];
    const float4* xi4 = (const float4*)(xb + i * C_);
#pragma unroll
    for (int c = 0; c < 16; ++c) xi[c] = xi4[c];

    // Stage diff rows (first 64 edge channels): lane r handles neighbor r.
    {
      const int j = idx[(size_t)p * K_ + lane];
      const float4* xj4 = (const float4*)(xb + j * C_);
      float* row = &XN[wave][lane * XSTR];
#pragma unroll
      for (int c = 0; c < 16; ++c) {
        float4 q = xj4[c];
        float4 d;
        d.x = q.x - xi[c].x; d.y = q.y - xi[c].y;
        d.z = q.z - xi[c].z; d.w = q.w - xi[c].w;
        *(float4*)(row + c * 4) = d;
      }
    }
    __syncthreads();

    // ---- Layer 1: H1(32x64) = lrelu( E(32x128) * W1^T * s1 + b1 ) ----
#pragma unroll
    for (int mt = 0; mt < 2; ++mt) {
      const float* arow = &XN[wave][(mt * 16 + lo) * XSTR];
      v8f acc[4] = {};
#pragma unroll
      for (int kc = 0; kc < 32; ++kc) {
        v2f a;
        if (kc < 16) {  // diff half from LDS
          a = *(const v2f*)(arow + kc * 4 + 2 * hi);
        } else {        // center half: broadcast x_i from registers (row-invariant)
          float4 q = xi[kc - 16];
          a.x = hi ? q.z : q.x;
          a.y = hi ? q.w : q.y;
        }
        const v2f* bp = &W1P[(kc * 2 + hi) * 64 + lo];
        v2f b0 = bp[0], bv1 = bp[16], bv2 = bp[32], bv3 = bp[48];
        acc[0] = __builtin_amdgcn_wmma_f32_16x16x4_f32(false, a, false, b0,  (short)0, acc[0], false, false);
        acc[1] = __builtin_amdgcn_wmma_f32_16x16x4_f32(false, a, false, bv1, (short)0, acc[1], false, false);
        acc[2] = __builtin_amdgcn_wmma_f32_16x16x4_f32(false, a, false, bv2, (short)0, acc[2], false, false);
        acc[3] = __builtin_amdgcn_wmma_f32_16x16x4_f32(false, a, false, bv3, (short)0, acc[3], false, false);
      }
#pragma unroll
      for (int nt = 0; nt < 4; ++nt) {
        const int o = nt * 16 + lo;
        const float s = s1s[o], bb = b1s[o];
#pragma unroll
        for (int v = 0; v < 8; ++v) {  // C/D layout: lane holds col o, rows v + 8*hi
          float y = acc[nt][v] * s + bb;
          y = y > 0.f ? y : NEG_SLOPE * y;
          H1[wave][(mt * 16 + hi * 8 + v) * XSTR + o] = y;
        }
      }
    }
    __syncthreads();

    // ---- Layer 2 + fused max over k: out(128) = max_rows lrelu(H1*W2^T*s2+b2) ----
    float cm[8];
#pragma unroll
    for (int q = 0; q < 8; ++q) cm[q] = NEG_INF;
#pragma unroll
    for (int mt = 0; mt < 2; ++mt) {
      const float* arow = &H1[wave][(mt * 16 + lo) * XSTR];
#pragma unroll
      for (int ng = 0; ng < 2; ++ng) {
        v8f acc[4] = {};
#pragma unroll
        for (int kc = 0; kc < 16; ++kc) {
          v2f a = *(const v2f*)(arow + kc * 4 + 2 * hi);
          const v2f* bp = &W2P[(kc * 2 + hi) * 128 + ng * 64 + lo];
          v2f b0 = bp[0], bv1 = bp[16], bv2 = bp[32], bv3 = bp[48];
          acc[0] = __builtin_amdgcn_wmma_f32_16x16x4_f32(false, a, false, b0,  (short)0, acc[0], false, false);
          acc[1] = __builtin_amdgcn_wmma_f32_16x16x4_f32(false, a, false, bv1, (short)0, acc[1], false, false);
          acc[2] = __builtin_amdgcn_wmma_f32_16x16x4_f32(false, a, false, bv2, (short)0, acc[2], false, false);
          acc[3] = __builtin_amdgcn_wmma_f32_16x16x4_f32(false, a, false, bv3, (short)0, acc[3], false, false);
        }
#pragma unroll
        for (int nt = 0; nt < 4; ++nt) {
          const int o = ng * 64 + nt * 16 + lo;
          const float s = s2s[o], bb = b2s[o];
          float mx = NEG_INF;
#pragma unroll
          for (int v = 0; v < 8; ++v) {
            float y = acc[nt][v] * s + bb;
            y = y > 0.f ? y : NEG_SLOPE * y;
            mx = fmaxf(mx, y);
          }
          const int q = ng * 4 + nt;
          cm[q] = fmaxf(cm[q], mx);
        }
      }
    }
    // merge the two 8-row halves of each column (lanes l and l+16 share col)
#pragma unroll
    for (int q = 0; q < 8; ++q) cm[q] = fmaxf(cm[q], __shfl_xor(cm[q], 16, 32));
    if (lane < 16) {
      float* op = out + (size_t)p * O2_ + lo;
#pragma unroll
      for (int q = 0; q < 8; ++q) op[q * 16] = cm[q];
    }
    __syncthreads();
  }
}

extern "C" void kernel_launch(void* const* d_in, const int* in_sizes, int n_in,
                              void* d_out, int out_size, void* d_ws, size_t ws_size,
                              hipStream_t stream) {
  (void)in_sizes; (void)n_in; (void)out_size; (void)ws_size;
  const float* x  = (const float*)d_in[0];
  const float* W1 = (const float*)d_in[1];
  const float* g1 = (const float*)d_in[2];
  const float* b1 = (const float*)d_in[3];
  const float* W2 = (const float*)d_in[4];
  const float* g2 = (const float*)d_in[5];
  const float* b2 = (const float*)d_in[6];
  float* out = (float*)d_out;
  int* idx = (int*)d_ws;  // B*N*32 ints = 2 MB scratch

  knn_topk_kernel<<<(B_ * N_) / 4, 128, 0, stream>>>(x, idx);
  edgeconv_wmma_kernel<<<2048, 128, 0, stream>>>(x, idx, W1, g1, b1, W2, g2, b2, out);
}